// DenseEntangler_10582799417614
// MI455X (gfx1250) — compile-verified
//
#include <hip/hip_runtime.h>
#include <hip/hip_bf16.h>

// CDNA5 (gfx1250) DenseEntangler: 6 chained fp32 GEMMs (M=8192, N=K=1024)
// using V_WMMA_F32_16X16X4_F32. B tile stored K-pair-interleaved in LDS so
// each WMMA B fragment is one ds_load_b64 (no v_mov shuffles).

typedef __attribute__((ext_vector_type(2))) float v2f;
typedef __attribute__((ext_vector_type(8))) float v8f;

#define DLEG 32
#define NDIM 1024          // N = K = D*D
#define BM 128
#define BN 64
#define BK 32
#define ASTRIDE 36         // LDS row stride for As (16B-aligned rows, conflict-free reads)
#define BPAIR  160         // DWORD stride between K-pairs in Bs (%64==32 -> halves disjoint)

__global__ __launch_bounds__(256)
void entangler_gemm_f32(const float* __restrict__ src,
                        const float* __restrict__ node,  // this step's (D,D,D,D) tensor
                        const float* __restrict__ bias,
                        float* __restrict__ dst,
                        int first_step, int last_step)
{
    __shared__ float As[BM * ASTRIDE];
    __shared__ float Bs[(BK / 2) * BPAIR];   // (k,n) at [(k>>1)*BPAIR + n*2 + (k&1)]

    const int tid  = threadIdx.x;
    const int lane = tid & 31;
    const int wave = tid >> 5;
    const int hi   = (lane >> 4) & 1;   // high half-wave -> K offset +2 (A/B), M offset +8 (C/D)
    const int l16  = lane & 15;

    const int N0 = blockIdx.x * BN;
    const int R0 = blockIdx.y * BM;     // row r = b*32 + f ; R0 multiple of 128
    const int b0 = R0 >> 5;

    v8f acc0 = {}, acc1 = {}, acc2 = {}, acc3 = {};

    for (int v = 0; v < DLEG; ++v) {    // K-chunk c == leg value v (k = v*32 + u)
        // ---------------- stage A tile (BM x BK) ----------------
        if (first_step) {
            // A[r=(b,f)][k=(v,u)] = src[b*32768 + u*1024 + v*32 + f] ; contiguous in f
            const int f    = tid & 31;
            const int idx0 = tid >> 5;                 // 0..7
            #pragma unroll
            for (int p = 0; p < 16; ++p) {
                int lin = idx0 + p * 8;                // 0..127 -> (bb,u)
                int bb = lin >> 5, u = lin & 31;
                As[(bb * 32 + f) * ASTRIDE + u] =
                    src[(size_t)(b0 + bb) * 32768 + u * 1024 + v * 32 + f];
            }
        } else {
            // A[r=(b,f)][k=(v,u)] = src[b*32768 + v*1024 + f*32 + u] ; contiguous in u
            const int u4  = (tid & 7) * 4;             // 128-bit loads along u
            const int rr0 = tid >> 3;                  // 0..31
            #pragma unroll
            for (int p = 0; p < 4; ++p) {
                int rr = rr0 + p * 32;                 // 0..127
                int r  = R0 + rr;
                int b  = r >> 5, f = r & 31;
                const float4 a4 = *(const float4*)
                    (src + (size_t)b * 32768 + v * 1024 + f * 32 + u4);
                *(float4*)(&As[rr * ASTRIDE + u4]) = a4;
            }
        }
        // ---------------- stage B tile (BK x BN), K-pair interleaved ----------------
        {
            // B[k=(v,u)][n] = node[u*32768 + v*1024 + N0 + n] ; contiguous in n
            const int n4 = (tid & 15) * 4;
            const int u0 = tid >> 4;                   // 0..15
            #pragma unroll
            for (int p = 0; p < 2; ++p) {
                int u = u0 + p * 16;
                const float4 b4 = *(const float4*)
                    (node + (size_t)u * 32768 + v * 1024 + N0 + n4);
                float* bdst = &Bs[(u >> 1) * BPAIR + (u & 1)];
                bdst[(n4 + 0) * 2] = b4.x;
                bdst[(n4 + 1) * 2] = b4.y;
                bdst[(n4 + 2) * 2] = b4.z;
                bdst[(n4 + 3) * 2] = b4.w;
            }
        }
        __syncthreads();

        // ---------------- WMMA f32 16x16x4 over this K-chunk ----------------
        const int arow = wave * 16 + l16;              // A: M = lane&15
        const int bcol = l16 * 2;
        #pragma unroll
        for (int kk = 0; kk < BK; kk += 4) {
            const int kA = kk + hi * 2;                // VGPR j -> K = kA + j (kA even)
            v2f a;
            a.x = As[arow * ASTRIDE + kA];
            a.y = As[arow * ASTRIDE + kA + 1];

            const float* brow = &Bs[(kA >> 1) * BPAIR + bcol];
            v2f bf;
            bf = *(const v2f*)(brow + 0 * 32);
            acc0 = __builtin_amdgcn_wmma_f32_16x16x4_f32(false, a, false, bf,
                                                         (short)0, acc0, false, false);
            bf = *(const v2f*)(brow + 1 * 32);
            acc1 = __builtin_amdgcn_wmma_f32_16x16x4_f32(false, a, false, bf,
                                                         (short)0, acc1, false, false);
            bf = *(const v2f*)(brow + 2 * 32);
            acc2 = __builtin_amdgcn_wmma_f32_16x16x4_f32(false, a, false, bf,
                                                         (short)0, acc2, false, false);
            bf = *(const v2f*)(brow + 3 * 32);
            acc3 = __builtin_amdgcn_wmma_f32_16x16x4_f32(false, a, false, bf,
                                                         (short)0, acc3, false, false);
        }
        __syncthreads();
    }

    // ---------------- store C (row-major y[b,f,m,n]) ----------------
    const int rbase = R0 + wave * 16 + hi * 8;         // C/D: VGPR j -> M = j (+8 high half)
    #pragma unroll
    for (int nt = 0; nt < 4; ++nt) {
        const v8f acc = (nt == 0) ? acc0 : (nt == 1) ? acc1 : (nt == 2) ? acc2 : acc3;
        const int col = N0 + nt * 16 + l16;
        #pragma unroll
        for (int j = 0; j < 8; ++j) {
            const int row = rbase + j;
            float val = acc[j];
            if (last_step) {
                val += bias[(row & 31) * 1024 + col];
                val = fmaxf(val, 0.0f);
            }
            dst[(size_t)row * 1024 + col] = val;
        }
    }
}

extern "C" void kernel_launch(void* const* d_in, const int* in_sizes, int n_in,
                              void* d_out, int out_size, void* d_ws, size_t ws_size,
                              hipStream_t stream) {
    const float* in    = (const float*)d_in[0];   // (256, 32768) fp32
    const float* nodes = (const float*)d_in[1];   // (6, 32,32,32,32) fp32
    const float* bias  = (const float*)d_in[2];   // (32768,) fp32
    float* out = (float*)d_out;                   // (256, 32768) fp32
    float* ws  = (float*)d_ws;                    // needs 32 MB ping buffer

    dim3 grid(NDIM / BN, 8192 / BM);              // (16, 64)
    dim3 block(256);

    const float* cur = in;
    float* bufs[2] = { ws, out };                 // s0->ws, s1->out, s2->ws, s3->out, s4->ws
    for (int s = 0; s < 6; ++s) {
        float* dstp = (s == 5) ? out : bufs[s & 1];
        entangler_gemm_f32<<<grid, block, 0, stream>>>(
            cur, nodes + (size_t)s * 1048576, bias, dstp, (s == 0) ? 1 : 0, (s == 5) ? 1 : 0);
        cur = dstp;
    }
}